// SHOTDescriptor_75041668596326
// MI455X (gfx1250) — compile-verified
//
#include <hip/hip_runtime.h>
#include <hip/hip_bf16.h>

// SHOT descriptor for MI455X (gfx1250, wave32).
// Pipeline: x2 norms -> WMMA-based KNN(5) -> Jacobi 3x3 eigh -> binning.

#define BATCHES   8
#define NPTS      4096
#define NNEIGH    5
#define LBINS     10
#define TBINS     80

typedef __attribute__((ext_vector_type(2))) float v2f;
typedef __attribute__((ext_vector_type(4))) float v4f;
typedef __attribute__((ext_vector_type(8))) float v8f;

// ---------------------------------------------------------------- K0: norms
__global__ __launch_bounds__(256) void shot_x2_kernel(
    const float* __restrict__ pts, float* __restrict__ x2) {
  int i = blockIdx.x * blockDim.x + threadIdx.x;   // 0 .. B*N-1
  float x = pts[3 * i + 0];
  float y = pts[3 * i + 1];
  float z = pts[3 * i + 2];
  x2[i] = x * x + y * y + z * z;
}

// ---------------------------------------------------------------- K1: KNN
// One wave (32 threads) per 16-query tile. A = 16 candidates x 4 coords,
// B = 4 coords x 16 queries.  D[m][n] = dot(cand_m, query_n) in fp32.
// Wave32 f32 layouts (ISA 7.12.2):
//   A (16x4): lane<16 holds row M=lane, VGPR0=K0,VGPR1=K1; lane>=16 K2,K3.
//   B (4x16): lane<16 holds col N=lane, VGPR0=K0,VGPR1=K1; lane>=16 K2,K3.
//   D: lane<16 -> D[r][lane] r=0..7 ; lane>=16 -> D[r+8][lane-16].
__global__ __launch_bounds__(32) void shot_knn_kernel(
    const float* __restrict__ pts, const float* __restrict__ x2,
    int* __restrict__ knn_idx) {
  const int lane  = threadIdx.x;          // 0..31
  const int tile  = blockIdx.x;           // 0 .. B*(N/16)-1
  const int b     = tile / (NPTS / 16);
  const int qbase = (tile % (NPTS / 16)) * 16;
  const float* P  = pts + (size_t)b * NPTS * 3;
  const float* X2 = x2 + (size_t)b * NPTS;

  const int  lq = lane & 15;
  const bool hi = lane >= 16;

  // B operand: query tile (held constant for whole kernel).
  v2f breg;
  {
    const int q = qbase + lq;
    float b0 = P[3 * q + 0];
    float b1 = P[3 * q + 1];
    float b2 = P[3 * q + 2];
    breg.x = hi ? b2 : b0;
    breg.y = hi ? 0.0f : b1;
  }

  // Per-lane sorted (ascending) top-5 of key = |c|^2 - 2 c.q  (order == d2).
  float dist[NNEIGH];
  int   indx[NNEIGH];
#pragma unroll
  for (int k = 0; k < NNEIGH; ++k) { dist[k] = 3.0e38f; indx[k] = 0x7fffffff; }

  for (int cbase = 0; cbase < NPTS; cbase += 16) {
    // A operand: candidate tile.
    v2f areg;
    {
      const int c = cbase + lq;
      float a0 = P[3 * c + 0];
      float a1 = P[3 * c + 1];
      float a2 = P[3 * c + 2];
      areg.x = hi ? a2 : a0;
      areg.y = hi ? 0.0f : a1;
    }
    v8f cz = {0.f, 0.f, 0.f, 0.f, 0.f, 0.f, 0.f, 0.f};
    // (neg_a, A, neg_b, B, c_mod, C, reuse_a, reuse_b)
    v8f d = __builtin_amdgcn_wmma_f32_16x16x4_f32(
        false, areg, false, breg, (short)0, cz, false, false);

    const int mofs = hi ? 8 : 0;
    const v4f* xv = (const v4f*)(X2 + cbase + mofs);   // 32B aligned
    v4f xa = xv[0], xb = xv[1];
    float xr[8] = {xa.x, xa.y, xa.z, xa.w, xb.x, xb.y, xb.z, xb.w};

#pragma unroll
    for (int r = 0; r < 8; ++r) {
      const float key = xr[r] - 2.0f * d[r];
      const int   ci  = cbase + mofs + r;
      if (key < dist[NNEIGH - 1]) {
        // stable insertion position (ties keep earlier index first)
        int pos = 0;
#pragma unroll
        for (int s = 0; s < NNEIGH; ++s) pos += (dist[s] <= key) ? 1 : 0;
#pragma unroll
        for (int s = NNEIGH - 1; s > 0; --s) {
          bool shift = (s > pos);
          bool here  = (s == pos);
          dist[s] = shift ? dist[s - 1] : (here ? key : dist[s]);
          indx[s] = shift ? indx[s - 1] : (here ? ci : indx[s]);
        }
        if (pos == 0) { dist[0] = key; indx[0] = ci; }
      }
    }
  }

  // Merge the two half-lists (lane q has M=0..7 stripes, lane q+16 M=8..15).
  float cd[2 * NNEIGH];
  int   cix[2 * NNEIGH];
#pragma unroll
  for (int k = 0; k < NNEIGH; ++k) {
    cd[k]           = dist[k];
    cix[k]          = indx[k];
    cd[NNEIGH + k]  = __shfl_xor(dist[k], 16, 32);
    cix[NNEIGH + k] = __shfl_xor(indx[k], 16, 32);
  }
  if (!hi) {
    // rank-based select (static indices; (key, idx) lexicographic order)
    int rank[2 * NNEIGH];
#pragma unroll
    for (int e = 0; e < 2 * NNEIGH; ++e) {
      int rk = 0;
#pragma unroll
      for (int f = 0; f < 2 * NNEIGH; ++f) {
        if (f != e) {
          bool less = (cd[f] < cd[e]) || (cd[f] == cd[e] && cix[f] < cix[e]);
          rk += less ? 1 : 0;
        }
      }
      rank[e] = rk;
    }
    int* outp = knn_idx + ((size_t)b * NPTS + (qbase + lq)) * NNEIGH;
#pragma unroll
    for (int k = 0; k < NNEIGH; ++k) {
      int sel = 0;
#pragma unroll
      for (int e = 0; e < 2 * NNEIGH; ++e)
        if (rank[e] == k) sel = cix[e];
      outp[k] = sel;
    }
  }
}

// ---------------------------------------------------------------- K2: eigh
template <int P, int Q>
__device__ __forceinline__ void jacobi_rot(float a[3][3], float v[3][3]) {
  const float apq = a[P][Q];
  if (fabsf(apq) > 1e-30f) {
    const float app = a[P][P], aqq = a[Q][Q];
    const float theta = (aqq - app) / (2.0f * apq);
    const float sgn = (theta >= 0.0f) ? 1.0f : -1.0f;
    const float t = sgn / (fabsf(theta) + sqrtf(theta * theta + 1.0f));
    const float c = 1.0f / sqrtf(t * t + 1.0f);
    const float s = t * c;
    a[P][P] = app - t * apq;
    a[Q][Q] = aqq + t * apq;
    a[P][Q] = 0.0f;
    a[Q][P] = 0.0f;
    constexpr int R = 3 - P - Q;
    const float arp = a[R][P], arq = a[R][Q];
    a[R][P] = c * arp - s * arq; a[P][R] = a[R][P];
    a[R][Q] = s * arp + c * arq; a[Q][R] = a[R][Q];
#pragma unroll
    for (int i = 0; i < 3; ++i) {
      const float vip = v[i][P], viq = v[i][Q];
      v[i][P] = c * vip - s * viq;
      v[i][Q] = s * vip + c * viq;
    }
  }
}

__global__ __launch_bounds__(256) void shot_lrf_kernel(
    const float* __restrict__ pts, const int* __restrict__ knn_idx,
    float* __restrict__ lrfs, float* __restrict__ normals) {
  const int p = blockIdx.x * blockDim.x + threadIdx.x;   // 0..B*N-1
  const int b = p / NPTS;
  const float* P = pts + (size_t)b * NPTS * 3;
  const int* nb = knn_idx + (size_t)p * NNEIGH;

  float nx[NNEIGH], ny[NNEIGH], nz[NNEIGH];
  float mx = 0.f, my = 0.f, mz = 0.f;
#pragma unroll
  for (int k = 0; k < NNEIGH; ++k) {
    const int j = nb[k];
    nx[k] = P[3 * j + 0]; ny[k] = P[3 * j + 1]; nz[k] = P[3 * j + 2];
    mx += nx[k]; my += ny[k]; mz += nz[k];
  }
  mx *= (1.0f / NNEIGH); my *= (1.0f / NNEIGH); mz *= (1.0f / NNEIGH);

  float a[3][3] = {{0.f,0.f,0.f},{0.f,0.f,0.f},{0.f,0.f,0.f}};
#pragma unroll
  for (int k = 0; k < NNEIGH; ++k) {
    const float dx = nx[k] - mx, dy = ny[k] - my, dz = nz[k] - mz;
    a[0][0] += dx * dx; a[0][1] += dx * dy; a[0][2] += dx * dz;
    a[1][1] += dy * dy; a[1][2] += dy * dz; a[2][2] += dz * dz;
  }
  a[1][0] = a[0][1]; a[2][0] = a[0][2]; a[2][1] = a[1][2];

  float v[3][3] = {{1.f,0.f,0.f},{0.f,1.f,0.f},{0.f,0.f,1.f}};
#pragma unroll
  for (int sweep = 0; sweep < 10; ++sweep) {
    jacobi_rot<0, 1>(a, v);
    jacobi_rot<0, 2>(a, v);
    jacobi_rot<1, 2>(a, v);
  }

  float e0 = a[0][0], e1 = a[1][1], e2 = a[2][2];
  float c0x = v[0][0], c0y = v[1][0], c0z = v[2][0];
  float c1x = v[0][1], c1y = v[1][1], c1z = v[2][1];
  float c2x = v[0][2], c2y = v[1][2], c2z = v[2][2];
  float t;
  if (e0 > e1) { t=e0;e0=e1;e1=t; t=c0x;c0x=c1x;c1x=t; t=c0y;c0y=c1y;c1y=t; t=c0z;c0z=c1z;c1z=t; }
  if (e1 > e2) { t=e1;e1=e2;e2=t; t=c1x;c1x=c2x;c2x=t; t=c1y;c1y=c2y;c2y=t; t=c1z;c1z=c2z;c2z=t; }
  if (e0 > e1) { t=e0;e0=e1;e1=t; t=c0x;c0x=c1x;c1x=t; t=c0y;c0y=c1y;c1y=t; t=c0z;c0z=c1z;c1z=t; }

  float* L = lrfs + (size_t)p * 9;   // row-major [i][j], columns = eigvecs asc
  L[0] = c0x; L[1] = c1x; L[2] = c2x;
  L[3] = c0y; L[4] = c1y; L[5] = c2y;
  L[6] = c0z; L[7] = c1z; L[8] = c2z;
  float* Nrm = normals + (size_t)p * 3;
  Nrm[0] = c0x; Nrm[1] = c0y; Nrm[2] = c0z;
}

// ---------------------------------------------------------------- K3: bins
__global__ __launch_bounds__(256) void shot_desc_kernel(
    const float* __restrict__ pts, const int* __restrict__ knn_idx,
    const float* __restrict__ lrfs, const float* __restrict__ normals,
    float* __restrict__ out) {
  const int p = blockIdx.x * blockDim.x + threadIdx.x;   // 0..B*N-1
  const int b = p / NPTS;
  const float* P = pts + (size_t)b * NPTS * 3;
  const int* nb = knn_idx + (size_t)p * NNEIGH;
  const float* L = lrfs + (size_t)p * 9;

  const float l00 = L[0], l01 = L[1], l02 = L[2];
  const float l10 = L[3], l11 = L[4], l12 = L[5];
  const float l20 = L[6], l21 = L[7], l22 = L[8];
  const float n0x = l00, n0y = l10, n0z = l20;   // own normal = column 0

  int bin[NNEIGH];
#pragma unroll
  for (int k = 0; k < NNEIGH; ++k) {
    const int j = nb[k];
    const float x = P[3 * j + 0], y = P[3 * j + 1], z = P[3 * j + 2];
    // proj = nbh(row) . lrf(columns)   (absolute coords, per reference)
    const float p0 = x * l00 + y * l10 + z * l20;
    const float p1 = x * l01 + y * l11 + z * l21;
    const float p2 = x * l02 + y * l12 + z * l22;
    const int spatial = ((p0 >= 0.f) ? 4 : 0) + ((p1 >= 0.f) ? 2 : 0) +
                        ((p2 >= 0.f) ? 1 : 0);
    const float* Nn = normals + ((size_t)b * NPTS + j) * 3;
    const float cosv = n0x * Nn[0] + n0y * Nn[1] + n0z * Nn[2];
    int nid = (int)floorf((float)LBINS * (cosv + 1.0f) * 0.5f);
    nid = nid < 0 ? 0 : (nid > LBINS - 1 ? LBINS - 1 : nid);
    bin[k] = spatial * LBINS + nid;
  }

  float* orow = out + (size_t)p * TBINS;
#pragma unroll 4
  for (int j = 0; j < TBINS; ++j) {
    float cnt = 0.f;
#pragma unroll
    for (int k = 0; k < NNEIGH; ++k) cnt += (bin[k] == j) ? 1.0f : 0.0f;
    orow[j] = cnt;
  }
}

// ---------------------------------------------------------------- launcher
extern "C" void kernel_launch(void* const* d_in, const int* in_sizes, int n_in,
                              void* d_out, int out_size, void* d_ws, size_t ws_size,
                              hipStream_t stream) {
  (void)in_sizes; (void)n_in; (void)out_size; (void)ws_size;
  const float* pts = (const float*)d_in[0];
  float* out = (float*)d_out;

  char* ws = (char*)d_ws;
  const size_t NP = (size_t)BATCHES * NPTS;
  float* x2      = (float*)ws;                              // NP floats
  int*   knn     = (int*)(ws + NP * 4);                     // NP*5 ints
  float* lrfs    = (float*)(ws + NP * 4 + NP * NNEIGH * 4); // NP*9 floats
  float* normals = (float*)(ws + NP * 4 + NP * NNEIGH * 4 + NP * 9 * 4);

  shot_x2_kernel<<<(int)(NP / 256), 256, 0, stream>>>(pts, x2);
  shot_knn_kernel<<<BATCHES * (NPTS / 16), 32, 0, stream>>>(pts, x2, knn);
  shot_lrf_kernel<<<(int)(NP / 256), 256, 0, stream>>>(pts, knn, lrfs, normals);
  shot_desc_kernel<<<(int)(NP / 256), 256, 0, stream>>>(pts, knn, lrfs, normals, out);
}